// CBAM_82033875354009
// MI455X (gfx1250) — compile-verified
//
#include <hip/hip_runtime.h>
#include <hip/hip_bf16.h>
#include <math.h>

// ---------------------------------------------------------------------------
// CBAM forward, MI455X (gfx1250).
// Bandwidth-bound: x = 102.8 MB fits in 192 MB L2, so the 3 passes that read x
// cost ~1 HBM read total; HBM traffic ~ 206 MB -> ~9 us roofline @ 23.3 TB/s.
// The tiny channel-MLP runs as exact-f32 WMMA (v_wmma_f32_16x16x4_f32) in a
// single wave32 workgroup (EXEC all ones, as WMMA requires).
// ---------------------------------------------------------------------------

#define B_  32
#define C_  256
#define H_  56
#define W_  56
#define HW_ (H_ * W_)
#define MID_ 16

typedef __attribute__((ext_vector_type(2))) float v2f;
typedef __attribute__((ext_vector_type(8))) float v8f;

// ---------------------------------------------------------------------------
// Pass 1: per-(b,c) spatial mean & max.  8192 blocks x 256 thr, reads x once.
// ---------------------------------------------------------------------------
__global__ __launch_bounds__(256) void cbam_pool_kernel(
    const float* __restrict__ x, float* __restrict__ avg_p, float* __restrict__ max_p) {
  const int bc = blockIdx.x;                  // b*C + c
  const float* xp = x + (size_t)bc * HW_;
  const int tid = threadIdx.x;
  float s = 0.f, mx = -3.402823466e38f;
  for (int i = tid; i < HW_; i += 256) {
    float v = xp[i];
    s += v;
    mx = fmaxf(mx, v);
  }
  __shared__ float ss[256];
  __shared__ float sm[256];
  ss[tid] = s; sm[tid] = mx;
  __syncthreads();
  for (int o = 128; o > 0; o >>= 1) {
    if (tid < o) {
      ss[tid] += ss[tid + o];
      sm[tid] = fmaxf(sm[tid], sm[tid + o]);
    }
    __syncthreads();
  }
  if (tid == 0) {
    avg_p[bc] = ss[0] * (1.0f / (float)HW_);
    max_p[bc] = sm[0];
  }
}

// ---------------------------------------------------------------------------
// Pass 2: channel-attention MLP via f32 WMMA, single wave32.
//   P[64,256] = [avg_p; max_p];  H = tanh(bn1(P @ w1^T));  O = tanh(bn2(H @ w2^T))
//   ca = sigmoid(O[0:32] + O[32:64])
// A-frag (16x4 f32): lane m = l&15, k-pair = (l>>4)*2.  B-frag mirrored.
// C/D (16x16 f32, 8 VGPR): row = r + 8*(l>=16), col = l&15.
// ---------------------------------------------------------------------------
__global__ __launch_bounds__(32) void cbam_mlp_ca_kernel(
    const float* __restrict__ avg_p, const float* __restrict__ max_p,
    const float* __restrict__ w1,
    const float* __restrict__ g1, const float* __restrict__ b1,
    const float* __restrict__ m1, const float* __restrict__ v1,
    const float* __restrict__ w2,
    const float* __restrict__ g2, const float* __restrict__ b2,
    const float* __restrict__ m2, const float* __restrict__ v2,
    float* __restrict__ ca) {
  __shared__ float Hs[64][17];               // padded to dodge bank conflicts

  const int lane = threadIdx.x;              // 0..31, one full wave32
  const int m    = lane & 15;                // A row / B col / D col for this lane
  const int kb   = (lane >> 4) * 2;          // K sub-pair held by this half-wave
  const int rofs = (lane >> 4) * 8;          // D row offset for this half-wave

  // ---- GEMM1: [64,256] x [256,16] -> Hs[64][16] ----
  // per-lane BN1 constants (column = m)
  const float s1 = g1[m] * rsqrtf(v1[m] + 1e-5f);
  const float o1 = b1[m] - m1[m] * s1;

  for (int mt = 0; mt < 4; ++mt) {
    const float* Pm = (mt < 2) ? (avg_p + (size_t)(mt * 16) * C_)
                               : (max_p + (size_t)((mt - 2) * 16) * C_);
    v8f acc = {0.f, 0.f, 0.f, 0.f, 0.f, 0.f, 0.f, 0.f};
    for (int k0 = 0; k0 < C_; k0 += 4) {
      v2f a, b;
      a.x = Pm[(size_t)m * C_ + k0 + kb];
      a.y = Pm[(size_t)m * C_ + k0 + kb + 1];
      b.x = w1[(size_t)m * C_ + k0 + kb];      // w1[n][k], n == lane&15
      b.y = w1[(size_t)m * C_ + k0 + kb + 1];
      acc = __builtin_amdgcn_wmma_f32_16x16x4_f32(
          /*neg_a=*/false, a, /*neg_b=*/false, b,
          /*c_mod=*/(short)0, acc, /*reuse_a=*/false, /*reuse_b=*/false);
    }
#pragma unroll
    for (int r = 0; r < 8; ++r) {
      const int row = mt * 16 + r + rofs;
      Hs[row][m] = tanhf(acc[r] * s1 + o1);
    }
  }
  __syncthreads();

  // ---- GEMM2: [64,16] x [16,256]; fuse avg(rows mt) + max(rows mt+2) ----
  for (int mt = 0; mt < 2; ++mt) {
    for (int nt = 0; nt < 16; ++nt) {
      v8f acc_a = {0.f, 0.f, 0.f, 0.f, 0.f, 0.f, 0.f, 0.f};
      v8f acc_b = {0.f, 0.f, 0.f, 0.f, 0.f, 0.f, 0.f, 0.f};
      const int n = m;                       // column within N-tile
      const int c = nt * 16 + n;
#pragma unroll
      for (int k0 = 0; k0 < MID_; k0 += 4) {
        v2f aa, ab, b;
        aa.x = Hs[mt * 16 + m][k0 + kb];
        aa.y = Hs[mt * 16 + m][k0 + kb + 1];
        ab.x = Hs[(mt + 2) * 16 + m][k0 + kb];
        ab.y = Hs[(mt + 2) * 16 + m][k0 + kb + 1];
        b.x  = w2[(size_t)c * MID_ + k0 + kb];   // w2[n][k]
        b.y  = w2[(size_t)c * MID_ + k0 + kb + 1];
        acc_a = __builtin_amdgcn_wmma_f32_16x16x4_f32(
            false, aa, false, b, (short)0, acc_a, false, false);
        acc_b = __builtin_amdgcn_wmma_f32_16x16x4_f32(
            false, ab, false, b, (short)0, acc_b, false, false);
      }
      const float s2 = g2[c] * rsqrtf(v2[c] + 1e-5f);
      const float o2 = b2[c] - m2[c] * s2;
#pragma unroll
      for (int r = 0; r < 8; ++r) {
        const float va  = tanhf(acc_a[r] * s2 + o2);
        const float vb  = tanhf(acc_b[r] * s2 + o2);
        const float sum = va + vb;
        const float sig = 1.0f / (1.0f + expf(-sum));
        const int brow  = mt * 16 + r + rofs;    // batch index 0..31
        ca[brow * C_ + c] = sig;
      }
    }
  }
}

// ---------------------------------------------------------------------------
// Pass 3: per-(b,h,w) channel mean & max of x*ca.  Re-reads x (L2 hit).
// ---------------------------------------------------------------------------
__global__ __launch_bounds__(256) void cbam_spatial_stats_kernel(
    const float* __restrict__ x, const float* __restrict__ ca,
    float* __restrict__ avgc, float* __restrict__ maxc) {
  const int t = blockIdx.x * 256 + threadIdx.x;
  if (t >= B_ * HW_) return;
  const int b = t / HW_;
  const int hw = t - b * HW_;
  const float* xb  = x + (size_t)b * C_ * HW_ + hw;
  const float* cab = ca + b * C_;
  float s = 0.f, mx = -3.402823466e38f;
  for (int c = 0; c < C_; ++c) {
    const float v = xb[(size_t)c * HW_] * cab[c];
    s += v;
    mx = fmaxf(mx, v);
  }
  avgc[t] = s * (1.0f / (float)C_);
  maxc[t] = mx;
}

// ---------------------------------------------------------------------------
// Pass 4: 3x3 conv (reflect pad) on [avg_c; max_c] + 2 BNs + sigmoid -> sa.
// ---------------------------------------------------------------------------
__device__ __forceinline__ int cbam_refl(int p, int n) {
  return p < 0 ? -p : (p >= n ? 2 * n - 2 - p : p);
}

__global__ __launch_bounds__(256) void cbam_sconv_kernel(
    const float* __restrict__ avgc, const float* __restrict__ maxc,
    const float* __restrict__ wgt, const float* __restrict__ bias,
    const float* __restrict__ sg1, const float* __restrict__ sb1,
    const float* __restrict__ sm1, const float* __restrict__ sv1,
    const float* __restrict__ sg2, const float* __restrict__ sb2,
    const float* __restrict__ sm2, const float* __restrict__ sv2,
    float* __restrict__ sa) {
  const int t = blockIdx.x * 256 + threadIdx.x;
  if (t >= B_ * HW_) return;
  const int b = t / HW_;
  const int hw = t - b * HW_;
  const int h = hw / W_;
  const int w = hw - h * W_;
  float acc = bias[0];
#pragma unroll
  for (int dh = -1; dh <= 1; ++dh) {
#pragma unroll
    for (int dw = -1; dw <= 1; ++dw) {
      const int hh = cbam_refl(h + dh, H_);
      const int ww = cbam_refl(w + dw, W_);
      const int o  = b * HW_ + hh * W_ + ww;
      const int wi = (dh + 1) * 3 + (dw + 1);
      acc += avgc[o] * wgt[wi] + maxc[o] * wgt[9 + wi];  // in-ch 0 = avg, 1 = max
    }
  }
  const float c1 = sg1[0] * rsqrtf(sv1[0] + 1e-3f);      // BasicConv2d BN (eps 1e-3)
  acc = (acc - sm1[0]) * c1 + sb1[0];
  const float c2 = sg2[0] * rsqrtf(sv2[0] + 1e-5f);      // SpatialAttention BN
  acc = (acc - sm2[0]) * c2 + sb2[0];
  sa[t] = 1.0f / (1.0f + expf(-acc));
}

// ---------------------------------------------------------------------------
// Pass 5: out = relu(fbn(x*(1 + ca*sa))).  float4 along W, reads x from L2.
// ---------------------------------------------------------------------------
__global__ __launch_bounds__(256) void cbam_final_kernel(
    const float* __restrict__ x, const float* __restrict__ ca,
    const float* __restrict__ sa,
    const float* __restrict__ fg, const float* __restrict__ fb,
    const float* __restrict__ fm, const float* __restrict__ fv,
    float* __restrict__ out) {
  const size_t t = (size_t)blockIdx.x * 256 + threadIdx.x;
  const size_t i = t * 4;
  if (i >= (size_t)B_ * C_ * HW_) return;
  const int b  = (int)(i / ((size_t)C_ * HW_));
  const int c  = (int)((i / HW_) % C_);
  const int hw = (int)(i % HW_);             // HW_ % 4 == 0 -> same (b,c) for all 4
  const float4 xv = *(const float4*)(x + i);
  const float4 sv = *(const float4*)(sa + (size_t)b * HW_ + hw);
  const float cav = ca[b * C_ + c];
  const float sf  = fg[c] * rsqrtf(fv[c] + 1e-5f);
  const float bf  = fb[c] - fm[c] * sf;
  float4 o;
  o.x = fmaxf(0.f, (xv.x * (1.0f + cav * sv.x)) * sf + bf);
  o.y = fmaxf(0.f, (xv.y * (1.0f + cav * sv.y)) * sf + bf);
  o.z = fmaxf(0.f, (xv.z * (1.0f + cav * sv.z)) * sf + bf);
  o.w = fmaxf(0.f, (xv.w * (1.0f + cav * sv.w)) * sf + bf);
  *(float4*)(out + i) = o;
}

// ---------------------------------------------------------------------------
extern "C" void kernel_launch(void* const* d_in, const int* in_sizes, int n_in,
                              void* d_out, int out_size, void* d_ws, size_t ws_size,
                              hipStream_t stream) {
  (void)in_sizes; (void)n_in; (void)out_size; (void)ws_size;

  const float* x       = (const float*)d_in[0];
  const float* w1      = (const float*)d_in[1];
  const float* bn1_g   = (const float*)d_in[2];
  const float* bn1_b   = (const float*)d_in[3];
  const float* bn1_m   = (const float*)d_in[4];
  const float* bn1_v   = (const float*)d_in[5];
  const float* w2      = (const float*)d_in[6];
  const float* bn2_g   = (const float*)d_in[7];
  const float* bn2_b   = (const float*)d_in[8];
  const float* bn2_m   = (const float*)d_in[9];
  const float* bn2_v   = (const float*)d_in[10];
  const float* sconv_w = (const float*)d_in[11];
  const float* sconv_b = (const float*)d_in[12];
  const float* sbn1_g  = (const float*)d_in[13];
  const float* sbn1_b  = (const float*)d_in[14];
  const float* sbn1_m  = (const float*)d_in[15];
  const float* sbn1_v  = (const float*)d_in[16];
  const float* sbn2_g  = (const float*)d_in[17];
  const float* sbn2_b  = (const float*)d_in[18];
  const float* sbn2_m  = (const float*)d_in[19];
  const float* sbn2_v  = (const float*)d_in[20];
  const float* fbn_g   = (const float*)d_in[21];
  const float* fbn_b   = (const float*)d_in[22];
  const float* fbn_m   = (const float*)d_in[23];
  const float* fbn_v   = (const float*)d_in[24];
  float* out = (float*)d_out;

  // workspace layout (floats)
  float* ws    = (float*)d_ws;
  float* avg_p = ws;                       // [B*C]      = 8192
  float* max_p = avg_p + B_ * C_;          // [B*C]      = 8192
  float* ca    = max_p + B_ * C_;          // [B*C]      = 8192
  float* avgc  = ca + B_ * C_;             // [B*HW]     = 100352
  float* maxc  = avgc + B_ * HW_;          // [B*HW]     = 100352
  float* sa    = maxc + B_ * HW_;          // [B*HW]     = 100352

  // 1) spatial pools
  cbam_pool_kernel<<<B_ * C_, 256, 0, stream>>>(x, avg_p, max_p);

  // 2) channel-attention MLP (single wave32, WMMA f32)
  cbam_mlp_ca_kernel<<<1, 32, 0, stream>>>(
      avg_p, max_p, w1, bn1_g, bn1_b, bn1_m, bn1_v,
      w2, bn2_g, bn2_b, bn2_m, bn2_v, ca);

  // 3) channel stats of x*ca
  const int np = B_ * HW_;
  cbam_spatial_stats_kernel<<<(np + 255) / 256, 256, 0, stream>>>(x, ca, avgc, maxc);

  // 4) spatial-attention conv + BNs + sigmoid
  cbam_sconv_kernel<<<(np + 255) / 256, 256, 0, stream>>>(
      avgc, maxc, sconv_w, sconv_b,
      sbn1_g, sbn1_b, sbn1_m, sbn1_v,
      sbn2_g, sbn2_b, sbn2_m, sbn2_v, sa);

  // 5) fused scale + residual + BN + ReLU
  const size_t ntot4 = (size_t)B_ * C_ * HW_ / 4;
  cbam_final_kernel<<<(unsigned)((ntot4 + 255) / 256), 256, 0, stream>>>(
      x, ca, sa, fbn_g, fbn_b, fbn_m, fbn_v, out);
}